// Deblur_route_19782619365725
// MI455X (gfx1250) — compile-verified
//
#include <hip/hip_runtime.h>
#include <cstddef>

typedef __attribute__((ext_vector_type(2))) float v2f;
typedef __attribute__((ext_vector_type(8))) float v8f;

#define H_   256
#define W_   256
#define NB_  2
#define CIN_ 64
#define DIM_ 32
#define HW_  ((size_t)H_ * (size_t)W_)

static __device__ __forceinline__ v8f wmma_f32(v2f a, v2f b, v8f c) {
  // D(16x16) = A(16x4, f32) * B(4x16, f32) + C ; emits v_wmma_f32_16x16x4_f32
  return __builtin_amdgcn_wmma_f32_16x16x4_f32(false, a, false, b, (short)0, c,
                                               false, false);
}

static __device__ __forceinline__ float selu_f(float v) {
  return 1.0507009873554805f *
         (v > 0.f ? v : 1.6732632423543772f * (__expf(v) - 1.f));
}

static __device__ __forceinline__ int clampi(int v, int lo, int hi) {
  return v < lo ? lo : (v > hi ? hi : v);
}

// ---------------------------------------------------------------------------
// One-shot reorder of w1 (OIHW: [oc][ci][tap]) -> w1t ([tap][oc][ci]) so each
// WMMA A-fragment K-pair (ci, ci+1) is contiguous: one global_load_b64 lands
// it directly in an even-aligned VGPR pair (no v_mov packing before v_wmma).
// ---------------------------------------------------------------------------
__global__ __launch_bounds__(256) void k_reorder_w1(
    const float* __restrict__ w1, float* __restrict__ w1t) {
  const int idx = blockIdx.x * 256 + threadIdx.x;  // < 32*64*9 = 18432
  const int tap = idx % 9;
  const int ci = (idx / 9) % CIN_;
  const int oc = idx / (9 * CIN_);
  w1t[((size_t)tap * DIM_ + oc) * CIN_ + ci] = w1[idx];
}

// ---------------------------------------------------------------------------
// conv1: (N,64,H,W) -> (N,32,H,W), 3x3 zero-pad, via WMMA f32 16x16x4.
// Wave: 16 pixels of one row x 32 output channels (2 M-tiles).
// A = reordered weights (M=oc, K=ci; K-pair contiguous -> b64 loads),
// B = input (K=ci, N=pixel). Zero padding is branch-free: coordinates are
// clamped in-bounds (loads always legal) and the value selected to 0 via
// v_cndmask — no exec diamonds in the hot loop.
// ---------------------------------------------------------------------------
__global__ __launch_bounds__(256) void k_conv1_wmma(
    const float* __restrict__ in, const float* __restrict__ w1t,
    const float* __restrict__ b1, float* __restrict__ out) {
  const int lane = threadIdx.x & 31;
  const int wv = threadIdx.x >> 5;
  const int m = lane & 15;
  const int hi = (lane >> 4) & 1;
  const int y = blockIdx.y;
  const int n = blockIdx.z;
  const int px0 = blockIdx.x * 128 + wv * 16;

  v8f acc0 = {};  // oc 0..15
  v8f acc1 = {};  // oc 16..31

  for (int dy = -1; dy <= 1; ++dy) {
    const int yy = y + dy;
    const bool yok = (unsigned)yy < (unsigned)H_;
    const int yc = clampi(yy, 0, H_ - 1);
    #pragma unroll
    for (int dx = -1; dx <= 1; ++dx) {
      const int xx = px0 + m + dx;
      const bool ok = yok && ((unsigned)xx < (unsigned)W_);
      const int xc = clampi(xx, 0, W_ - 1);
      const int tap = (dy + 1) * 3 + (dx + 1);
      const float* inb =
          in + (size_t)(n * CIN_) * HW_ + (size_t)yc * W_ + xc;
      // Per-lane A base rows for this tap (oc = m and oc = m+16).
      const float* wT = w1t + (size_t)tap * DIM_ * CIN_;
      const float* wA0 = wT + (size_t)m * CIN_;
      const float* wA1 = wT + (size_t)(m + 16) * CIN_;
      #pragma unroll
      for (int c = 0; c < CIN_; c += 4) {
        // B fragment: V0 rows K=c (lo lanes) / K=c+1 (hi lanes); V1 rows +2
        const int kb = c + hi;
        const float v0 = inb[(size_t)kb * HW_];
        const float v1 = inb[(size_t)(kb + 2) * HW_];
        v2f bfr;
        bfr.x = ok ? v0 : 0.f;  // v_cndmask, straight-line zero padding
        bfr.y = ok ? v1 : 0.f;
        // A fragments: V0 cols K=c (lo) / K=c+2 (hi); V1 cols K=c+1 / K=c+3.
        // K-pair contiguous in w1t -> single b64 load each (ka even, 8B align)
        const int ka = c + 2 * hi;
        const v2f a0 = *(const v2f*)(wA0 + ka);
        const v2f a1 = *(const v2f*)(wA1 + ka);
        acc0 = wmma_f32(a0, bfr, acc0);
        acc1 = wmma_f32(a1, bfr, acc1);
      }
    }
  }
  // D layout: VGPR j holds row M=j (lanes 0-15) / M=j+8 (lanes 16-31), N=lane%16
  #pragma unroll
  for (int j = 0; j < 8; ++j) {
    const int oc = j + 8 * hi;
    out[(size_t)(n * DIM_ + oc) * HW_ + (size_t)y * W_ + px0 + m] =
        acc0[j] + b1[oc];
    out[(size_t)(n * DIM_ + 16 + oc) * HW_ + (size_t)y * W_ + px0 + m] =
        acc1[j] + b1[16 + oc];
  }
}

// ---------------------------------------------------------------------------
// conv2: (N,32,H,W) -> (N,4,H,W), 3x3 zero-pad + SELU. Tiny: plain VALU.
// ---------------------------------------------------------------------------
__global__ __launch_bounds__(256) void k_conv2_selu(
    const float* __restrict__ x1, const float* __restrict__ w2,
    const float* __restrict__ b2, float* __restrict__ x2) {
  const int pix = blockIdx.x * 256 + threadIdx.x;
  const int n = blockIdx.y;
  const int y = pix >> 8;
  const int x = pix & (W_ - 1);
  float acc[4];
  #pragma unroll
  for (int o = 0; o < 4; ++o) acc[o] = b2[o];
  for (int ci = 0; ci < DIM_; ++ci) {
    const float* src = x1 + (size_t)(n * DIM_ + ci) * HW_;
    #pragma unroll
    for (int dy = -1; dy <= 1; ++dy) {
      const int yy = y + dy;
      if ((unsigned)yy >= (unsigned)H_) continue;
      #pragma unroll
      for (int dx = -1; dx <= 1; ++dx) {
        const int xx = x + dx;
        if ((unsigned)xx >= (unsigned)W_) continue;
        const float v = src[(size_t)yy * W_ + xx];
        const int t = (dy + 1) * 3 + (dx + 1);
        #pragma unroll
        for (int o = 0; o < 4; ++o)
          acc[o] = fmaf(v, w2[(size_t)(o * DIM_ + ci) * 9 + t], acc[o]);
      }
    }
  }
  #pragma unroll
  for (int o = 0; o < 4; ++o)
    x2[(size_t)(n * 4 + o) * HW_ + pix] = selu_f(acc[o]);
}

// ---------------------------------------------------------------------------
// conv3: (N,4,H,W) -> (N,32,H,W), 3x3 zero-pad + SELU. Tiny: plain VALU.
// ---------------------------------------------------------------------------
__global__ __launch_bounds__(256) void k_conv3_selu(
    const float* __restrict__ x2, const float* __restrict__ w3,
    const float* __restrict__ b3, float* __restrict__ x3) {
  const int pix = blockIdx.x * 256 + threadIdx.x;
  const int n = blockIdx.y;
  const int y = pix >> 8;
  const int x = pix & (W_ - 1);
  float acc[DIM_];
  #pragma unroll
  for (int o = 0; o < DIM_; ++o) acc[o] = b3[o];
  #pragma unroll
  for (int ci = 0; ci < 4; ++ci) {
    const float* src = x2 + (size_t)(n * 4 + ci) * HW_;
    #pragma unroll
    for (int dy = -1; dy <= 1; ++dy) {
      const int yy = y + dy;
      if ((unsigned)yy >= (unsigned)H_) continue;
      #pragma unroll
      for (int dx = -1; dx <= 1; ++dx) {
        const int xx = x + dx;
        if ((unsigned)xx >= (unsigned)W_) continue;
        const float v = src[(size_t)yy * W_ + xx];
        const int t = (dy + 1) * 3 + (dx + 1);
        #pragma unroll
        for (int o = 0; o < DIM_; ++o)
          acc[o] = fmaf(v, w3[(size_t)(o * 4 + ci) * 9 + t], acc[o]);
      }
    }
  }
  #pragma unroll
  for (int o = 0; o < DIM_; ++o)
    x3[(size_t)(n * DIM_ + o) * HW_ + pix] = selu_f(acc[o]);
}

// ---------------------------------------------------------------------------
// Fused conv4 (1x1, 32->800) + per-pixel 5x5 blur (edge-replicate pad).
// Wave: 32-pixel strip of one row. x3 strip (B operands) loaded ONCE into
// registers and reused for all 32 output channels. Per channel: 32 WMMAs
// produce the 25 (padded to 32) per-pixel kernel values; bias b4 folded in at
// LDS-spill time; then 25 FMAs per lane/pixel. w4 A-fragments are contiguous
// K-pairs -> single b64 loads, branch-free masking for the padded M-tile rows.
// The 419 MB blur_kernel tensor never touches HBM.
// ---------------------------------------------------------------------------
__global__ __launch_bounds__(256) void k_conv4_blur_wmma(
    const float* __restrict__ x3, const float* __restrict__ w4,
    const float* __restrict__ b4, const float* __restrict__ img,
    float* __restrict__ out) {
  __shared__ float kbuf[8][32][33];  // [wave][kernel-row][pixel] (+pad)
  const int lane = threadIdx.x & 31;
  const int wv = threadIdx.x >> 5;
  const int m = lane & 15;
  const int hi = (lane >> 4) & 1;
  const int y = blockIdx.x;
  const int n = blockIdx.y;
  const int px0 = wv * 32;

  // B fragments: x3 strip, (K=4ci x N=16px) per chunk, 2 strips of 16 px.
  v2f Bf[2][8];
  #pragma unroll
  for (int s = 0; s < 2; ++s) {
    const int px = px0 + s * 16 + m;
    const float* src = x3 + (size_t)(n * DIM_) * HW_ + (size_t)y * W_ + px;
    #pragma unroll
    for (int c = 0; c < 8; ++c) {
      const int kb = 4 * c + hi;
      Bf[s][c].x = src[(size_t)kb * HW_];
      Bf[s][c].y = src[(size_t)(kb + 2) * HW_];
    }
  }

  const float msk1 = (m < 9) ? 1.f : 0.f;   // rows 25..31 of Mtile1 = padding
  const int mr = (m < 9) ? m : 8;           // clamp keeps w4 loads in-bounds

  for (int oc = 0; oc < DIM_; ++oc) {
    if (oc + 1 < DIM_)  // stream next channel's w4 rows (global_prefetch_b8)
      __builtin_prefetch(w4 + (size_t)(oc + 1) * 25 * DIM_, 0, 1);

    v8f acc[2][2] = {};  // [Mtile (kernel rows 0-15 / 16-31)][strip]
    const float* wR0 = w4 + (size_t)(oc * 25 + m) * DIM_;
    const float* wR1 = w4 + (size_t)(oc * 25 + 16 + mr) * DIM_;
    #pragma unroll
    for (int c = 0; c < 8; ++c) {
      const int ka = 4 * c + 2 * hi;        // even -> 8B-aligned b64 loads
      const v2f a0 = *(const v2f*)(wR0 + ka);
      v2f a1 = *(const v2f*)(wR1 + ka);
      a1.x *= msk1;  // branch-free zeroing of padded rows
      a1.y *= msk1;
      #pragma unroll
      for (int s = 0; s < 2; ++s) {
        acc[0][s] = wmma_f32(a0, Bf[s][c], acc[0][s]);
        acc[1][s] = wmma_f32(a1, Bf[s][c], acc[1][s]);
      }
    }

    // Spill K-tile (32 rows x 32 px) to per-wave LDS, folding in bias b4.
    // Rows >= 25 are padding; guard keeps b4 reads in-bounds (800 entries).
    #pragma unroll
    for (int j = 0; j < 8; ++j) {
      const int r = j + 8 * hi;
      const float bias0 = b4[oc * 25 + r];  // r in 0..15, always valid
      const float bias1 = (16 + r < 25) ? b4[oc * 25 + 16 + r] : 0.f;
      #pragma unroll
      for (int s = 0; s < 2; ++s) {
        kbuf[wv][r][s * 16 + m] = acc[0][s][j] + bias0;
        kbuf[wv][16 + r][s * 16 + m] = acc[1][s][j] + bias1;
      }
    }
    __syncthreads();

    // Stage 2: per-pixel 5x5 blur, edge-replicate clamping.
    const int myx = px0 + lane;
    float sum = 0.f;
    #pragma unroll
    for (int a = 0; a < 5; ++a) {
      const int yy = clampi(y + a - 2, 0, H_ - 1);
      const float* srow =
          img + (size_t)(n * DIM_ + oc) * HW_ + (size_t)yy * W_;
      #pragma unroll
      for (int b = 0; b < 5; ++b) {
        const int xx = clampi(myx + b - 2, 0, W_ - 1);
        sum = fmaf(srow[xx], kbuf[wv][a * 5 + b][lane], sum);
      }
    }
    out[(size_t)(n * DIM_ + oc) * HW_ + (size_t)y * W_ + myx] = sum;
    __syncthreads();
  }
}

extern "C" void kernel_launch(void* const* d_in, const int* in_sizes, int n_in,
                              void* d_out, int out_size, void* d_ws,
                              size_t ws_size, hipStream_t stream) {
  (void)in_sizes; (void)n_in; (void)out_size; (void)ws_size;
  const float* img  = (const float*)d_in[0];  // (2,32,256,256)
  const float* dmap = (const float*)d_in[1];  // (2,64,256,256)
  const float* w1 = (const float*)d_in[2];
  const float* b1 = (const float*)d_in[3];
  const float* w2 = (const float*)d_in[4];
  const float* b2 = (const float*)d_in[5];
  const float* w3 = (const float*)d_in[6];
  const float* b3 = (const float*)d_in[7];
  const float* w4 = (const float*)d_in[8];
  const float* b4 = (const float*)d_in[9];
  float* out = (float*)d_out;

  float* x1  = (float*)d_ws;                        // (2,32,256,256)
  float* x2  = x1 + (size_t)NB_ * DIM_ * HW_;       // (2, 4,256,256)
  float* x3  = x2 + (size_t)NB_ * 4 * HW_;          // (2,32,256,256)
  float* w1t = x3 + (size_t)NB_ * DIM_ * HW_;       // 9*32*64 floats

  k_reorder_w1<<<dim3((DIM_ * CIN_ * 9) / 256), 256, 0, stream>>>(w1, w1t);
  k_conv1_wmma<<<dim3(W_ / 128, H_, NB_), 256, 0, stream>>>(dmap, w1t, b1, x1);
  k_conv2_selu<<<dim3(H_ * W_ / 256, NB_), 256, 0, stream>>>(x1, w2, b2, x2);
  k_conv3_selu<<<dim3(H_ * W_ / 256, NB_), 256, 0, stream>>>(x2, w3, b3, x3);
  k_conv4_blur_wmma<<<dim3(H_, NB_), 256, 0, stream>>>(x3, w4, b4, img, out);
}